// MoDRouter_39316130627985
// MI455X (gfx1250) — compile-verified
//
#include <hip/hip_runtime.h>
#include <hip/hip_bf16.h>

// MoD router for MI455X (gfx1250, wave32).
// Bandwidth-bound mat-vec done with full-fp32 V_WMMA_F32_16X16X4_F32,
// then an LDS radix-select top-k and a deterministic aux-loss reduction.

typedef __attribute__((ext_vector_type(2))) float v2f;
typedef __attribute__((ext_vector_type(8))) float v8f;

#define D_MODEL 2048
#define SEQ_L   8192
#define BATCH   4
#define BL      (BATCH * SEQ_L)   // 32768
#define CAP     4096              // SEQ_L * 0.5
#define ROWS_PER_BLOCK 128        // 8 waves * 16 rows
#define NBLOCKS (BL / ROWS_PER_BLOCK)  // 256 (64 blocks per batch row)

// Map float -> uint32 so unsigned compare == float compare (total order).
__device__ __forceinline__ unsigned order_key(float f) {
  unsigned u = __float_as_uint(f);
  return (u & 0x80000000u) ? ~u : (u | 0x80000000u);
}

// ---------------------------------------------------------------------------
// Kernel 1: logits = x @ w via WMMA f32 16x16x4, plus noisy keys + sigmoid
// partial sums. Each wave: 16 rows (M), loop K over D in chunks of 4.
// A layout (32-bit A 16x4): lanes 0-15 hold {K0,K1}, lanes 16-31 hold {K2,K3},
// M = lane & 15. B (4x16) replicated across N: VGPR0/1 hold {K0,K1} for lanes
// 0-15 and {K2,K3} for lanes 16-31 (mirrors the halving pattern of C/D).
// C/D layout: column N=0 lives in lane 0 (M=0..7, VGPR0..7) and lane 16
// (M=8..15, VGPR0..7).
// ---------------------------------------------------------------------------
__global__ __launch_bounds__(256) void mod_router_matvec(
    const float* __restrict__ x, const float* __restrict__ noise,
    const float* __restrict__ w, float* __restrict__ logits_out,
    unsigned* __restrict__ keys, float* __restrict__ partial) {
  __shared__ float sw[D_MODEL];    // 8 KB staged copy of w
  __shared__ float wave_sig[16];   // per (wave, half) sigmoid partials

  const int tid = threadIdx.x;
#pragma unroll
  for (int i = 0; i < D_MODEL / 256; ++i) sw[tid + i * 256] = w[tid + i * 256];
  __syncthreads();

  const int lane = tid & 31;
  const int wave = tid >> 5;
  const int row0 = blockIdx.x * ROWS_PER_BLOCK + wave * 16;  // global (b*L + l)
  const int half = lane >> 4;      // 0: K0/K1 holder, 1: K2/K3 holder
  const int koff = half << 1;
  const float* xrow = x + (size_t)(row0 + (lane & 15)) * D_MODEL + koff;

  v8f c0 = {};
  v8f c1 = {};
  for (int k0 = 0; k0 < D_MODEL; k0 += 8) {
    v2f a0 = *(const v2f*)(xrow + k0);
    v2f b0 = *(const v2f*)(&sw[k0 + koff]);
    c0 = __builtin_amdgcn_wmma_f32_16x16x4_f32(false, a0, false, b0,
                                               (short)0, c0, false, false);
    v2f a1 = *(const v2f*)(xrow + k0 + 4);
    v2f b1 = *(const v2f*)(&sw[k0 + 4 + koff]);
    c1 = __builtin_amdgcn_wmma_f32_16x16x4_f32(false, a1, false, b1,
                                               (short)0, c1, false, false);
  }
  v8f c = c0 + c1;

  // Column 0 extraction: lanes 0 (M=0..7) and 16 (M=8..15).
  if ((lane & 15) == 0) {
    const int mbase = row0 + half * 8;
    float sig = 0.f;
#pragma unroll
    for (int j = 0; j < 8; ++j) {
      float v = c[j];
      logits_out[mbase + j] = v;
      keys[mbase + j] = order_key(fmaf(noise[mbase + j], 0.1f, v));
      sig += 1.0f / (1.0f + expf(-v));
    }
    wave_sig[wave * 2 + half] = sig;
  }
  __syncthreads();
  if (tid == 0) {  // deterministic sequential block reduction
    float s = 0.f;
#pragma unroll
    for (int i = 0; i < 16; ++i) s += wave_sig[i];
    partial[blockIdx.x] = s;
  }
}

// ---------------------------------------------------------------------------
// Kernel 2: per batch row, exact top-4096 threshold via 4-pass radix select
// over order-keys in LDS; ties broken deterministically by lowest index
// (matches jax.lax.top_k). Emits 0/1 float mask.
// ---------------------------------------------------------------------------
__global__ __launch_bounds__(256) void mod_router_select(
    const unsigned* __restrict__ keys, float* __restrict__ mask_out) {
  __shared__ unsigned skeys[SEQ_L];  // 32 KB
  __shared__ unsigned hist[256];
  __shared__ unsigned cnts[256];
  __shared__ unsigned s_prefix;
  __shared__ int s_k;

  const int b = blockIdx.x;
  const int tid = threadIdx.x;
  const unsigned* kb = keys + b * SEQ_L;

#pragma unroll
  for (int i = 0; i < SEQ_L / 256; ++i) skeys[tid + i * 256] = kb[tid + i * 256];
  if (tid == 0) { s_prefix = 0u; s_k = CAP; }
  __syncthreads();

  for (int shift = 24; shift >= 0; shift -= 8) {
    hist[tid] = 0u;
    __syncthreads();
    const unsigned pref = s_prefix;
    const unsigned prefmask = (shift == 24) ? 0u : (0xFFFFFFFFu << (shift + 8));
    for (int i = 0; i < SEQ_L / 256; ++i) {
      unsigned key = skeys[tid + i * 256];
      if ((key & prefmask) == pref)
        atomicAdd(&hist[(key >> shift) & 255u], 1u);
    }
    __syncthreads();
    if (tid == 0) {
      int k = s_k;
      int cum = 0;
      int bin = 255;
      for (; bin >= 0; --bin) {
        cum += (int)hist[bin];
        if (cum >= k) break;
      }
      int above = cum - (int)hist[bin];  // strictly-greater count this level
      s_k = k - above;                   // remaining picks inside chosen bin
      s_prefix = pref | ((unsigned)bin << shift);
    }
    __syncthreads();
  }

  const unsigned T = s_prefix;  // exact 4096th-largest key
  const int need = s_k;         // how many key==T elements to keep

  // Deterministic index-ordered rank of ties: contiguous 32-element chunks.
  const int base = tid * (SEQ_L / 256);
  int cnt = 0;
#pragma unroll
  for (int i = 0; i < SEQ_L / 256; ++i) cnt += (skeys[base + i] == T) ? 1 : 0;
  cnts[tid] = (unsigned)cnt;
  __syncthreads();
  if (tid == 0) {  // exclusive scan, sequential => deterministic
    unsigned run = 0;
    for (int i = 0; i < 256; ++i) { unsigned c = cnts[i]; cnts[i] = run; run += c; }
  }
  __syncthreads();

  int rank = (int)cnts[tid];
  float* mout = mask_out + b * SEQ_L;
  for (int i = 0; i < SEQ_L / 256; ++i) {
    unsigned key = skeys[base + i];
    float m = 0.f;
    if (key > T) {
      m = 1.f;
    } else if (key == T) {
      m = (rank < need) ? 1.f : 0.f;
      ++rank;
    }
    mout[base + i] = m;
  }
}

// ---------------------------------------------------------------------------
// Kernel 3: aux loss from per-block sigmoid partials (sequential, exact).
// ---------------------------------------------------------------------------
__global__ void mod_router_aux(const float* __restrict__ partial,
                               float* __restrict__ aux_out) {
  if (threadIdx.x == 0) {
    float acc = 0.f;
    for (int b = 0; b < BATCH; ++b) {
      float s = 0.f;
      for (int i = 0; i < NBLOCKS / BATCH; ++i)
        s += partial[b * (NBLOCKS / BATCH) + i];
      float m = s * (1.0f / SEQ_L) - 0.5f;
      acc += m * m;
    }
    aux_out[0] = 0.01f * (acc * (1.0f / BATCH));
  }
}

extern "C" void kernel_launch(void* const* d_in, const int* in_sizes, int n_in,
                              void* d_out, int out_size, void* d_ws, size_t ws_size,
                              hipStream_t stream) {
  (void)in_sizes; (void)n_in; (void)out_size; (void)ws_size;
  const float* x     = (const float*)d_in[0];  // [B, L, D]
  const float* noise = (const float*)d_in[1];  // [B, L]
  const float* w     = (const float*)d_in[2];  // [D]

  float* out        = (float*)d_out;
  float* mask_out   = out;            // [B*L] 0/1 mask
  float* logits_out = out + BL;       // [B*L] logits
  float* aux_out    = out + 2 * BL;   // [1]   aux loss

  unsigned* keys  = (unsigned*)d_ws;        // B*L order-keys (128 KB)
  float*    partial = (float*)d_ws + BL;    // NBLOCKS sigmoid partials

  mod_router_matvec<<<NBLOCKS, 256, 0, stream>>>(x, noise, w, logits_out, keys, partial);
  mod_router_select<<<BATCH, 256, 0, stream>>>(keys, mask_out);
  mod_router_aux<<<1, 64, 0, stream>>>(partial, aux_out);
}